// ForgetMult_23948737642764
// MI455X (gfx1250) — compile-verified
//
#include <hip/hip_runtime.h>
#include <stdint.h>

// ForgetMult scan: h_t = f_t*x_t + (1-f_t)*h_{t-1}  == h + f*(x - h)
// [T, B, H] layout, channels c = b*H + h are fully independent.
// Bandwidth-bound (384 MB traffic, ~2 flop/elem) -> pipeline memory with
// CDNA5 async global->LDS loads (ASYNCcnt), triple-buffered per wave so the
// 6-bit ASYNCcnt never saturates (max 48 outstanding < 63).

#define K_CHUNK 8                 // timesteps staged per chunk
#define NBUF 3                    // triple buffering (prefetch depth 2 chunks)
#define BLOCK_T 256               // threads per block
#define WPB (BLOCK_T / 32)        // waves per block (wave32)

#if defined(__HIP_DEVICE_COMPILE__)

#if __has_builtin(__builtin_amdgcn_s_wait_asynccnt)
#define WAIT_ASYNC(n) __builtin_amdgcn_s_wait_asynccnt(n)
#else
#define WAIT_ASYNC(n) asm volatile("s_wait_asynccnt %0" ::"n"(n) : "memory")
#endif

// Truncate a generic (shared-aperture) pointer to its 32-bit LDS offset.
__device__ __forceinline__ unsigned lds_off(const void* p) {
  return (unsigned)(uintptr_t)p;
}

// Issue async global->LDS b32 loads for one K_CHUNK x 32-lane tile of f and x.
// SADDR form: global_load_async_to_lds_b32 vdst(ldsoff), vaddr(byteoff), s[base].
__device__ __forceinline__ void issue_chunk(
    float (&lf)[NBUF][WPB][K_CHUNK][32], float (&lx)[NBUF][WPB][K_CHUNK][32],
    const float* __restrict__ f, const float* __restrict__ x,
    int buf, int t0, unsigned BH, unsigned c, int wave, int lane) {
#pragma unroll
  for (int k = 0; k < K_CHUNK; ++k) {
    unsigned voff = ((unsigned)(t0 + k) * BH + c) * 4u;
    unsigned df = lds_off(&lf[buf][wave][k][lane]);
    unsigned dx = lds_off(&lx[buf][wave][k][lane]);
    asm volatile("global_load_async_to_lds_b32 %0, %1, %2"
                 :: "v"(df), "v"(voff), "s"(f)
                 : "memory");
    asm volatile("global_load_async_to_lds_b32 %0, %1, %2"
                 :: "v"(dx), "v"(voff), "s"(x)
                 : "memory");
  }
}

#endif  // __HIP_DEVICE_COMPILE__

__global__ __launch_bounds__(BLOCK_T) void ForgetMult_async_kernel(
    const float* __restrict__ f, const float* __restrict__ x,
    const float* __restrict__ h0, float* __restrict__ out,
    int T, int BHi) {
#if defined(__HIP_DEVICE_COMPILE__)
  __shared__ float lf[NBUF][WPB][K_CHUNK][32];
  __shared__ float lx[NBUF][WPB][K_CHUNK][32];

  const int lane = threadIdx.x & 31;
  const int wave = threadIdx.x >> 5;
  const unsigned BH = (unsigned)BHi;
  const unsigned c = blockIdx.x * (unsigned)BLOCK_T + threadIdx.x;

  float h = h0[c];

  const int nchunk = T / K_CHUNK;

  // Prologue: stage chunks 0 and 1.
  issue_chunk(lf, lx, f, x, 0, 0, BH, c, wave, lane);
  if (nchunk > 1)
    issue_chunk(lf, lx, f, x, 1, K_CHUNK, BH, c, wave, lane);

  int buf = 0;
  for (int i = 0; i < nchunk; ++i) {
    if (i + 2 < nchunk) {
      // Stage chunk i+2, then wait until chunk i's loads (oldest in flight,
      // completing in order) have landed: two newer chunks = 4*K_CHUNK loads
      // may remain outstanding.
      int buf2 = buf + 2;
      if (buf2 >= NBUF) buf2 -= NBUF;
      issue_chunk(lf, lx, f, x, buf2, (i + 2) * K_CHUNK, BH, c, wave, lane);
      WAIT_ASYNC(4 * K_CHUNK);
    } else if (i + 1 < nchunk) {
      WAIT_ASYNC(2 * K_CHUNK);   // only chunk i+1 still in flight
    } else {
      WAIT_ASYNC(0);             // last chunk: drain everything
    }

    size_t base = (size_t)i * K_CHUNK * BH + c;
#pragma unroll
    for (int k = 0; k < K_CHUNK; ++k) {
      float fv = lf[buf][wave][k][lane];
      float xv = lx[buf][wave][k][lane];
      h = fmaf(fv, xv - h, h);                       // h + f*(x - h)
      __builtin_nontemporal_store(h, &out[base + (size_t)k * BH]);
    }

    if (++buf >= NBUF) buf -= NBUF;
  }
#endif
}

// Generic fallback (any shape), plain global loads.
__global__ void ForgetMult_simple_kernel(
    const float* __restrict__ f, const float* __restrict__ x,
    const float* __restrict__ h0, float* __restrict__ out,
    int T, long long BH) {
  long long c = (long long)blockIdx.x * blockDim.x + threadIdx.x;
  if (c >= BH) return;
  float h = h0[c];
  for (int t = 0; t < T; ++t) {
    float fv = f[(long long)t * BH + c];
    float xv = x[(long long)t * BH + c];
    h = fmaf(fv, xv - h, h);
    out[(long long)t * BH + c] = h;
  }
}

extern "C" void kernel_launch(void* const* d_in, const int* in_sizes, int n_in,
                              void* d_out, int out_size, void* d_ws, size_t ws_size,
                              hipStream_t stream) {
  (void)n_in; (void)out_size; (void)d_ws; (void)ws_size;
  const float* f  = (const float*)d_in[0];
  const float* x  = (const float*)d_in[1];
  const float* h0 = (const float*)d_in[2];
  float* out = (float*)d_out;

  const long long BH = in_sizes[2];                 // B*H
  const int T = (int)(in_sizes[0] / (BH > 0 ? BH : 1));

  const bool fast_ok =
      BH > 0 && T > 0 &&
      (BH % BLOCK_T) == 0 &&
      (T % K_CHUNK) == 0 &&
      ((long long)T * BH * 4ll) < (1ll << 31);      // 32-bit byte offsets

  if (fast_ok) {
    dim3 grid((unsigned)(BH / BLOCK_T));
    ForgetMult_async_kernel<<<grid, BLOCK_T, 0, stream>>>(f, x, h0, out, T, (int)BH);
  } else {
    int threads = 256;
    long long blocks = (BH + threads - 1) / threads;
    if (blocks < 1) blocks = 1;
    ForgetMult_simple_kernel<<<(unsigned)blocks, threads, 0, stream>>>(f, x, h0, out, T, BH);
  }
}